// GRPOLoss_8796093022768
// MI455X (gfx1250) — compile-verified
//
#include <hip/hip_runtime.h>
#include <stdint.h>
#include <math.h>

// Problem constants (match reference).
#define B_     2
#define L_     1024
#define V_     50257
#define BETA_  0.04f
#define EPS_LO 0.2f
#define EPS_HI 0.2f

// Tiling: 256 threads/block (8 wave32), 16 floats per thread per tile.
#define TPB        256
#define KCH        16
#define TILE_ELEMS (TPB * KCH)                      // 4096 floats / tile
#define NTILES     ((V_ + TILE_ELEMS - 1) / TILE_ELEMS)  // 13

// ---------------------------------------------------------------------------
// CDNA5 async global->LDS path (gfx1250). Guarded so the file always compiles;
// fallback is a plain register-tile streaming loop with identical math.
// ---------------------------------------------------------------------------
#if defined(__HIP_DEVICE_COMPILE__) && defined(__gfx1250__)
  #if __has_builtin(__builtin_amdgcn_global_load_async_to_lds_b32)
    #define ASYNC_LDS 1
  #else
    #define ASYNC_LDS 0
    #warning "gfx1250: __builtin_amdgcn_global_load_async_to_lds_b32 not found; using register-tile fallback"
  #endif
  #if __has_builtin(__builtin_amdgcn_s_wait_asynccnt)
    #define WAIT_ASYNC(n) do { __builtin_amdgcn_s_wait_asynccnt(n); asm volatile("" ::: "memory"); } while (0)
  #else
    #define WAIT_ASYNC(n) asm volatile("s_wait_asynccnt %0" :: "n"(n) : "memory")
  #endif
#else
  #define ASYNC_LDS 0
  #define WAIT_ASYNC(n) do {} while (0)
#endif

#if ASYNC_LDS
// Builtin parameters are int-typed pointers (global src, lds dst, imm, imm).
typedef __attribute__((address_space(1))) int gbl_i32_t;
typedef __attribute__((address_space(3))) int lds_i32_t;
__device__ __forceinline__ void async_ld_f32(float* lds_dst, const float* gsrc) {
  __builtin_amdgcn_global_load_async_to_lds_b32(
      (gbl_i32_t*)(uintptr_t)gsrc, (lds_i32_t*)lds_dst, 0, 0);
}
#endif

#define NEG_BIG (-1e30f)

// ---------------------------------------------------------------------------
// Kernel 1: per-row online logsumexp -> logp[row] into workspace.
// One block per (b, l) token position; row length V_.
// ---------------------------------------------------------------------------
__global__ __launch_bounds__(TPB) void grpo_logp_kernel(
    const float* __restrict__ logits,     // (B, L+1, V)
    const int*   __restrict__ cids,       // (B, L)
    float*       __restrict__ logp_out)   // (B*L,)
{
  const int r   = blockIdx.x;             // 0 .. B*L-1
  const int tid = threadIdx.x;
  const int b   = r / L_;
  const int l   = r - b * L_;
  const float* __restrict__ row = logits + (size_t)(b * (L_ + 1) + l) * (size_t)V_;

  // Token logit (one lane; tiny, hits L2).
  float tok = 0.0f;
  if (tid == 0) tok = row[cids[r]];

  float m = NEG_BIG;  // running max
  float s = 0.0f;     // running sum of exp(x - m)

#if ASYNC_LDS
  // Per-wave independent double buffer: thread tid only ever touches
  // slot [.][k][tid], which its own wave's async loads filled, so no
  // work-group barriers are needed inside the pipeline.
  __shared__ float tile[2][KCH][TPB];     // 32 KB

  auto issue_tile = [&](int T, int buf) {
    const float* base = row + (size_t)T * TILE_ELEMS + tid;
#pragma unroll
    for (int k = 0; k < KCH; ++k)
      async_ld_f32(&tile[buf][k][tid], base + k * TPB);
  };

  issue_tile(0, 0);
  issue_tile(1, 1);

  for (int T = 0; T < NTILES; ++T) {
    // Outstanding: tiles T (16 ops) and T+1 (16 ops). Wait until tile T's
    // 16 per-wave async ops completed (loads complete in order).
    if (T + 1 < NTILES) { WAIT_ASYNC(KCH); } else { WAIT_ASYNC(0); }

    const int buf = T & 1;
    float x[KCH];
#pragma unroll
    for (int k = 0; k < KCH; ++k) x[k] = tile[buf][k][tid];

    if (T == NTILES - 1) {                // mask tail past V_
#pragma unroll
      for (int k = 0; k < KCH; ++k) {
        const int e = T * TILE_ELEMS + k * TPB + tid;
        if (e >= V_) x[k] = NEG_BIG;
      }
    }

    // Online logsumexp: tile max, single rescale, one exp per element.
    float tm = x[0];
#pragma unroll
    for (int k = 1; k < KCH; ++k) tm = fmaxf(tm, x[k]);
    const float mn = fmaxf(m, tm);
    s *= __expf(m - mn);                  // exp(-1e30 - finite) -> 0, safe
    m = mn;
#pragma unroll
    for (int k = 0; k < KCH; ++k) s += __expf(x[k] - m);

    // x[] fully consumed above (compiler's dscnt wait precedes this point),
    // so buf may now be refilled.
    asm volatile("" ::: "memory");
    if (T + 2 < NTILES) issue_tile(T + 2, buf);
  }
#else
  // Fallback: stream straight from global into registers.
  for (int T = 0; T < NTILES; ++T) {
    float x[KCH];
    const float* base = row + (size_t)T * TILE_ELEMS + tid;
#pragma unroll
    for (int k = 0; k < KCH; ++k) {
      const int e = T * TILE_ELEMS + k * TPB + tid;
      x[k] = (e < V_) ? base[k * TPB] : NEG_BIG;
    }
    float tm = x[0];
#pragma unroll
    for (int k = 1; k < KCH; ++k) tm = fmaxf(tm, x[k]);
    const float mn = fmaxf(m, tm);
    s *= __expf(m - mn);
    m = mn;
#pragma unroll
    for (int k = 0; k < KCH; ++k) s += __expf(x[k] - m);
  }
#endif

  // Cross-thread (m, s) merge. Every thread saw valid elements in tile 0,
  // so all m are finite here.
  __shared__ float rm[TPB];
  __shared__ float rs[TPB];
  rm[tid] = m; rs[tid] = s;
  __syncthreads();
  for (int off = TPB / 2; off > 0; off >>= 1) {
    if (tid < off) {
      const float m2 = rm[tid + off], s2 = rs[tid + off];
      const float M = fmaxf(m, m2);
      s = s * __expf(m - M) + s2 * __expf(m2 - M);
      m = M;
      rm[tid] = m; rs[tid] = s;
    }
    __syncthreads();
  }

  if (tid == 0) {
    const float lse = m + logf(s);
    logp_out[r] = tok - lse;              // TEMPERATURE == 1.0
  }
}

// ---------------------------------------------------------------------------
// Kernel 2: 2048 per-token values -> (reduced_loss, kl_mean, clip_ratio).
// Single block; trivially cheap.
// ---------------------------------------------------------------------------
__global__ __launch_bounds__(TPB) void grpo_finalize_kernel(
    const float* __restrict__ logp,       // (B*L,) from workspace
    const float* __restrict__ adv,        // (B,)
    const float* __restrict__ old_logp,   // (B, L)
    const float* __restrict__ ref_logp,   // (B, L)
    const int*   __restrict__ cmask,      // (B, L)
    float*       __restrict__ out)        // 3 scalars
{
  const int tid = threadIdx.x;
  float mask_sum = 0.f, kl_sum = 0.f, clip_sum = 0.f;
  float loss0 = 0.f, loss1s = 0.f, len0 = 0.f, len1 = 0.f;

  for (int r = tid; r < B_ * L_; r += TPB) {
    const int   b    = r / L_;
    const float lp   = logp[r];
    const float a    = adv[b];
    const float c1   = __expf(lp - old_logp[r]);
    const float c2   = fminf(fmaxf(c1, 1.f - EPS_LO), 1.f + EPS_HI);
    const float l1   = c1 * a;
    const float l2   = c2 * a;
    float       ptl  = -fminf(l1, l2);
    const float diff = ref_logp[r] - lp;
    const float kl   = __expf(diff) - diff - 1.f;
    ptl += BETA_ * kl;
    const float mk = (float)cmask[r];

    mask_sum += mk;
    kl_sum   += kl * mk;
    const bool clipped = ((c1 < 1.f - EPS_LO) && (a < 0.f)) ||
                         ((c1 > 1.f + EPS_HI) && (a > 0.f));
    clip_sum += clipped ? mk : 0.f;
    if (b == 0) { loss0  += ptl * mk; len0 += mk; }
    else        { loss1s += ptl * mk; len1 += mk; }
  }

  __shared__ float red[7][TPB];
  red[0][tid] = mask_sum; red[1][tid] = kl_sum; red[2][tid] = clip_sum;
  red[3][tid] = loss0;    red[4][tid] = loss1s;
  red[5][tid] = len0;     red[6][tid] = len1;
  __syncthreads();
  for (int off = TPB / 2; off > 0; off >>= 1) {
    if (tid < off) {
#pragma unroll
      for (int k = 0; k < 7; ++k) red[k][tid] += red[k][tid + off];
    }
    __syncthreads();
  }

  if (tid == 0) {
    const float ms         = fmaxf(red[0][0], 1.f);
    const float kl_mean    = red[1][0] / ms;
    const float clip_ratio = red[2][0] / ms;
    const float rl         = 0.5f * (red[3][0] / fmaxf(red[5][0], 1.f) +
                                     red[4][0] / fmaxf(red[6][0], 1.f));
    out[0] = rl;
    out[1] = kl_mean;
    out[2] = clip_ratio;
  }
}

// ---------------------------------------------------------------------------
extern "C" void kernel_launch(void* const* d_in, const int* in_sizes, int n_in,
                              void* d_out, int out_size, void* d_ws, size_t ws_size,
                              hipStream_t stream) {
  (void)in_sizes; (void)n_in; (void)out_size; (void)ws_size;
  const float* logits   = (const float*)d_in[0];
  const int*   cids     = (const int*)  d_in[1];
  const float* adv      = (const float*)d_in[2];
  const float* old_logp = (const float*)d_in[3];
  const float* ref_logp = (const float*)d_in[4];
  const int*   cmask    = (const int*)  d_in[5];
  float*       wlogp    = (float*)d_ws;          // B*L floats = 8 KB scratch

  grpo_logp_kernel<<<B_ * L_, TPB, 0, stream>>>(logits, cids, wlogp);
  grpo_finalize_kernel<<<1, TPB, 0, stream>>>(wlogp, adv, old_logp, ref_logp,
                                              cmask, (float*)d_out);
}